// GraphAttentionModule_48129403519482
// MI455X (gfx1250) — compile-verified
//
#include <hip/hip_runtime.h>
#include <hip/hip_bf16.h>

// GAT layer for MI455X (gfx1250, wave32, WMMA f32 16x16x4).
// B=8, N=2048, IN_F=128, OUT_F=64, ALPHA=0.2

typedef float v2f __attribute__((ext_vector_type(2)));
typedef float v8f __attribute__((ext_vector_type(8)));

#define GAT_B     8
#define GAT_N     2048
#define GAT_INF   128
#define GAT_OUTF  64
#define GAT_ROWS  (GAT_B * GAT_N)      // 16384
#define ALPHA     0.2f

__device__ __forceinline__ float lrelu(float x) {
    // x>0 -> x ; x<0 -> 0.2x  ==  max(x, 0.2x)
    return fmaxf(x, ALPHA * x);
}

#define WMMA_F32(A, Bv, C) \
    __builtin_amdgcn_wmma_f32_16x16x4_f32(false, (A), false, (Bv), (short)0, (C), false, false)

// ---------------------------------------------------------------------------
// Kernel 1: x_w = inputs @ W   (16384x128 @ 128x64, fp32 WMMA 16x16x4)
// One wave handles a 16-row x 64-col slab (4 n-tiles), K loop in steps of 4.
// Grid: 1024 waves = 128 blocks x 256 threads.
// ---------------------------------------------------------------------------
__global__ void gat_gemm_xw(const float* __restrict__ inp,
                            const float* __restrict__ W,
                            float* __restrict__ xw) {
    const int lane = threadIdx.x & 31;
    const int wid  = threadIdx.x >> 5;
    const int tile = blockIdx.x * 8 + wid;      // 0..1023
    const int rowbase = tile * 16;              // over 16384 rows
    const int m    = lane & 15;
    const int koff = (lane >> 4) << 1;          // lanes 0-15 -> K 0,1 ; 16-31 -> K 2,3

    v8f acc0 = {}, acc1 = {}, acc2 = {}, acc3 = {};
    const float* arow = inp + (size_t)(rowbase + m) * GAT_INF;

    for (int k = 0; k < GAT_INF; k += 4) {
        v2f A = *(const v2f*)(arow + k + koff);                 // 8B aligned (koff even)
        const float* w0 = W + (size_t)(k + koff) * GAT_OUTF;    // row K=koff
        const float* w1 = w0 + GAT_OUTF;                        // row K=koff+1
        v2f B0 = { w0[ 0 + m], w1[ 0 + m] };
        v2f B1 = { w0[16 + m], w1[16 + m] };
        v2f B2 = { w0[32 + m], w1[32 + m] };
        v2f B3 = { w0[48 + m], w1[48 + m] };
        acc0 = WMMA_F32(A, B0, acc0);
        acc1 = WMMA_F32(A, B1, acc1);
        acc2 = WMMA_F32(A, B2, acc2);
        acc3 = WMMA_F32(A, B3, acc3);
    }

    // C/D layout: VGPR r, lanes 0-15 -> row r, N=lane ; lanes 16-31 -> row r+8
    const int rhi = (lane >> 4) << 3;
    for (int r = 0; r < 8; ++r) {
        const int row = rowbase + r + rhi;
        float* orow = xw + (size_t)row * GAT_OUTF + m;
        orow[ 0] = acc0[r];
        orow[16] = acc1[r];
        orow[32] = acc2[r];
        orow[48] = acc3[r];
    }
}

// ---------------------------------------------------------------------------
// Kernel 2: s1/s2 scores. h = [nodes_emb | x_w]; s1 = h.a1, s2 = h.a2
// One thread per node. Grid: 64 blocks x 256.
// ---------------------------------------------------------------------------
__global__ void gat_scores(const float* __restrict__ ne,
                           const float* __restrict__ xw,
                           const float* __restrict__ a,
                           float* __restrict__ s1,
                           float* __restrict__ s2) {
    const int idx = blockIdx.x * 256 + threadIdx.x;   // 0..16383
    const float* n = ne + (size_t)idx * GAT_OUTF;
    const float* x = xw + (size_t)idx * GAT_OUTF;
    float v1 = 0.f, v2 = 0.f;
    for (int f = 0; f < GAT_OUTF; ++f) {
        const float nv = n[f];
        const float xv = x[f];
        v1 += nv * a[f]       + xv * a[ 64 + f];
        v2 += nv * a[128 + f] + xv * a[192 + f];
    }
    s1[idx] = v1;
    s2[idx] = v2;
}

// ---------------------------------------------------------------------------
// Kernel 3: per-batch max of s2 (softmax stabilization; lrelu is monotone so
// rowmax_i = lrelu(s1_i + max_j s2_j)). Grid: 8 blocks x 256.
// ---------------------------------------------------------------------------
__global__ void gat_s2max(const float* __restrict__ s2,
                          float* __restrict__ s2max) {
    __shared__ float red[256];
    const int b = blockIdx.x;
    float mx = -3.4e38f;
    for (int t = threadIdx.x; t < GAT_N; t += 256)
        mx = fmaxf(mx, s2[b * GAT_N + t]);
    red[threadIdx.x] = mx;
    __syncthreads();
    for (int s = 128; s > 0; s >>= 1) {
        if (threadIdx.x < s)
            red[threadIdx.x] = fmaxf(red[threadIdx.x], red[threadIdx.x + s]);
        __syncthreads();
    }
    if (threadIdx.x == 0) s2max[b] = red[0];
}

// ---------------------------------------------------------------------------
// Kernel 4: attention + h_prime.
// One wave per (batch, 16-row tile). Phase 1: accumulate unnormalized
// p_ij into WMMA A-operand layout and 4 f32 accumulators (h_prime_unnorm)
// plus per-row sums. Then scale h_prime by 1/rowsum + ReLU. Phase 2:
// recompute p and write normalized attention with coalesced stores.
// Grid: 1024 waves = 128 blocks x 256 threads.
// ---------------------------------------------------------------------------
__global__ void gat_attn(const float* __restrict__ xw,
                         const float* __restrict__ s1g,
                         const float* __restrict__ s2g,
                         const float* __restrict__ s2maxg,
                         float* __restrict__ out_h,
                         float* __restrict__ out_att) {
    const int lane = threadIdx.x & 31;
    const int wid  = threadIdx.x >> 5;
    const int w    = blockIdx.x * 8 + wid;      // 0..1023
    const int b    = w >> 7;                    // /128
    const int it   = w & 127;
    const int rowbase = it * 16;
    const int m    = lane & 15;
    const int koff = (lane >> 4) << 1;

    const float* s2b = s2g + b * GAT_N;
    const float* xwb = xw + (size_t)b * GAT_N * GAT_OUTF;

    const float s1l  = s1g[b * GAT_N + rowbase + m];   // this lane's row score
    const float smax = s2maxg[b];
    const float mrow = lrelu(s1l + smax);              // row max of e

    float rs = 0.f;
    v8f acc0 = {}, acc1 = {}, acc2 = {}, acc3 = {};

    for (int jb = 0; jb < GAT_N; jb += 4) {
        const int j0 = jb + koff;
        const v2f s2p = *(const v2f*)(s2b + j0);       // broadcast within half-wave
        const float p0 = __expf(lrelu(s1l + s2p.x) - mrow);
        const float p1 = __expf(lrelu(s1l + s2p.y) - mrow);
        rs += p0 + p1;
        v2f A = { p0, p1 };
        const float* x0 = xwb + (size_t)j0 * GAT_OUTF; // row j0 of x_w
        const float* x1 = x0 + GAT_OUTF;               // row j0+1
        v2f B0 = { x0[ 0 + m], x1[ 0 + m] };
        v2f B1 = { x0[16 + m], x1[16 + m] };
        v2f B2 = { x0[32 + m], x1[32 + m] };
        v2f B3 = { x0[48 + m], x1[48 + m] };
        acc0 = WMMA_F32(A, B0, acc0);
        acc1 = WMMA_F32(A, B1, acc1);
        acc2 = WMMA_F32(A, B2, acc2);
        acc3 = WMMA_F32(A, B3, acc3);
    }

    // row m's total sum lives split across lanes m and m+16 (wave32)
    const float rtot = rs + __shfl_xor(rs, 16, 32);
    const float inv  = 1.0f / rtot;                    // lanes 0-15: inv for rows 0-15

    // scale + ReLU h_prime; accumulator row for VGPR r is r + 8*(lane>=16)
    const int rhi = (lane >> 4) << 3;
    for (int r = 0; r < 8; ++r) {
        const int row = r + rhi;
        const float invr = __shfl(inv, row, 32);       // ds_bpermute broadcast
        const size_t o = ((size_t)(b * GAT_N + rowbase + row)) * GAT_OUTF + m;
        out_h[o +  0] = fmaxf(acc0[r] * invr, 0.f);
        out_h[o + 16] = fmaxf(acc1[r] * invr, 0.f);
        out_h[o + 32] = fmaxf(acc2[r] * invr, 0.f);
        out_h[o + 48] = fmaxf(acc3[r] * invr, 0.f);
    }

    // Phase 2: write normalized attention, one row at a time, coalesced over j.
    for (int mr = 0; mr < 16; ++mr) {
        const float bs1  = __shfl(s1l,  mr, 32);
        const float bm   = __shfl(mrow, mr, 32);
        const float binv = __shfl(inv,  mr, 32);
        float* arow = out_att + ((size_t)(b * GAT_N + rowbase + mr)) * GAT_N;
        for (int j = lane; j < GAT_N; j += 32) {
            const float t = lrelu(bs1 + s2b[j]);
            arow[j] = __expf(t - bm) * binv;
        }
    }
}

// ---------------------------------------------------------------------------
extern "C" void kernel_launch(void* const* d_in, const int* in_sizes, int n_in,
                              void* d_out, int out_size, void* d_ws, size_t ws_size,
                              hipStream_t stream) {
    const float* inp = (const float*)d_in[0];   // (8,2048,128)
    const float* ne  = (const float*)d_in[1];   // (8,2048,64)
    const float* W   = (const float*)d_in[2];   // (128,64)
    const float* a   = (const float*)d_in[3];   // (256,1)

    float* out_h   = (float*)d_out;                         // 8*2048*64
    float* out_att = out_h + (size_t)GAT_ROWS * GAT_OUTF;   // 8*2048*2048

    float* xw  = (float*)d_ws;                   // 16384*64
    float* s1  = xw + (size_t)GAT_ROWS * GAT_OUTF;
    float* s2  = s1 + GAT_ROWS;
    float* s2m = s2 + GAT_ROWS;

    gat_gemm_xw<<<128, 256, 0, stream>>>(inp, W, xw);
    gat_scores <<< 64, 256, 0, stream>>>(ne, xw, a, s1, s2);
    gat_s2max  <<<  8, 256, 0, stream>>>(s2, s2m);
    gat_attn   <<<128, 256, 0, stream>>>(xw, s1, s2, s2m, out_h, out_att);
}